// SpatialFocus_69355131896093
// MI455X (gfx1250) — compile-verified
//
#include <hip/hip_runtime.h>
#include <math.h>

typedef __attribute__((ext_vector_type(2))) float v2f;
typedef __attribute__((ext_vector_type(8))) float v8f;

#define NSEG 32
#define PDIM 8
#define TDIM 8
#define FDIM 32
#define COLS 256            // T*F
#define ENC_NEG_INF 0x007FFFFFu

// Monotone order-preserving float<->uint encoding for atomicMax on floats.
__device__ __forceinline__ unsigned encf(float f) {
    unsigned u = __float_as_uint(f);
    return (u & 0x80000000u) ? ~u : (u | 0x80000000u);
}
__device__ __forceinline__ float decf(unsigned u) {
    return __uint_as_float((u & 0x80000000u) ? (u & 0x7FFFFFFFu) : ~u);
}

__device__ __forceinline__ v8f wmma_f32(v2f A, v2f B, v8f C) {
    return __builtin_amdgcn_wmma_f32_16x16x4_f32(
        /*neg_a=*/false, A, /*neg_b=*/false, B,
        /*c_mod=*/(short)0, C, /*reuse_a=*/false, /*reuse_b=*/false);
}

// -------- init workspace (runs every launch: deterministic start state) ------
__global__ void init_kernel(float* __restrict__ accum, unsigned* __restrict__ Mkey,
                            float* __restrict__ S) {
    int idx = blockIdx.x * blockDim.x + threadIdx.x;
    if (idx < NSEG * PDIM * COLS) accum[idx] = 0.0f;
    if (idx < NSEG * PDIM) { Mkey[idx] = ENC_NEG_INF; S[idx] = 0.0f; }
}

// -------- segment boundaries via binary search on sorted seg ----------------
__global__ void seg_starts_kernel(const int* __restrict__ seg, int N,
                                  int* __restrict__ starts) {
    int b = threadIdx.x;
    if (b > NSEG) return;
    int lo = 0, hi = N;
    while (lo < hi) { int mid = (lo + hi) >> 1; if (seg[mid] < b) lo = mid + 1; else hi = mid; }
    starts[b] = lo;
}

// -------- tiny MLP + per-segment max (LDS-staged atomics) -------------------
__global__ __launch_bounds__(256) void mlp_max_kernel(
    const float* __restrict__ pos, const int* __restrict__ seg,
    const float* __restrict__ W1, const float* __restrict__ b1,
    const float* __restrict__ W2, const float* __restrict__ b2,
    float* __restrict__ h_out, unsigned* __restrict__ Mkey, int N)
{
    __shared__ unsigned lmax[NSEG * PDIM];
    const int tid = threadIdx.x;
    lmax[tid] = ENC_NEG_INF;
    __syncthreads();
    const int n = blockIdx.x * 256 + tid;
    if (n < N) {
        const float p0 = pos[n*3+0], p1 = pos[n*3+1], p2 = pos[n*3+2];
        float hid[PDIM];
        #pragma unroll
        for (int p = 0; p < PDIM; ++p) {
            float v = p0*W1[0*PDIM+p] + p1*W1[1*PDIM+p] + p2*W1[2*PDIM+p] + b1[p];
            hid[p] = v > 0.0f ? v : expm1f(v);   // ELU(alpha=1)
        }
        const int s = seg[n];
        #pragma unroll
        for (int q = 0; q < PDIM; ++q) {
            float v = b2[q];
            #pragma unroll
            for (int p = 0; p < PDIM; ++p) v += hid[p] * W2[p*PDIM+q];
            h_out[n*PDIM + q] = v;               // TAU == 1.0 -> no divide
            atomicMax(&lmax[s*PDIM + q], encf(v));
        }
    }
    __syncthreads();
    const unsigned key = lmax[tid];
    if (key != ENC_NEG_INF) atomicMax(&Mkey[tid], key);
}

// -------- e = exp(h - max), per-segment sum (LDS-staged atomics) ------------
__global__ __launch_bounds__(256) void exp_sum_kernel(
    const int* __restrict__ seg, const unsigned* __restrict__ Mkey,
    float* __restrict__ e, float* __restrict__ S, int N)
{
    __shared__ float lsum[NSEG * PDIM];
    const int tid = threadIdx.x;
    lsum[tid] = 0.0f;
    __syncthreads();
    const int n = blockIdx.x * 256 + tid;
    if (n < N) {
        const int s = seg[n];
        #pragma unroll
        for (int q = 0; q < PDIM; ++q) {
            const float m = decf(Mkey[s*PDIM + q]);
            const float ev = __expf(e[n*PDIM + q] - m);
            e[n*PDIM + q] = ev;
            atomicAdd(&lsum[s*PDIM + q], ev);
        }
    }
    __syncthreads();
    const float v = lsum[tid];
    if (v != 0.0f) atomicAdd(&S[tid], v);
}

// -------- segmented GEMM via V_WMMA_F32_16X16X4_F32 -------------------------
// Per segment b: accum[b][p][col] += sum_n e[n][p] * x[n][col]   (col = t*32+f)
// M=16 (8 rows used, 8 zero-padded), N=16 per column tile, K=4 nodes per WMMA.
// grid = (32 segments, 2 coltile groups, 8 node chunks); 8 waves/block,
// wave w owns column tile blockIdx.y*8+w -> x is streamed exactly once.
// Main loop is branch-free (no exec masking): all loads unconditional,
// A-row padding via value select; one masked tail chunk handles remainder.
__global__ __launch_bounds__(256) void gemm_wmma_kernel(
    const float* __restrict__ x, const float* __restrict__ e,
    const int* __restrict__ starts, float* __restrict__ accum)
{
    const int wave = threadIdx.x >> 5;
    const int lane = threadIdx.x & 31;
    const int b = blockIdx.x;
    const int coltile = blockIdx.y * 8 + wave;   // 0..15
    const int col0 = coltile * 16;

    const int segStart = starts[b];
    const int segEnd   = starts[b + 1];
    const int len = segEnd - segStart;
    const int per = (len + (int)gridDim.z - 1) / (int)gridDim.z;
    const int c0 = segStart + (int)blockIdx.z * per;
    const int c1 = min(c0 + per, segEnd);
    if (c0 >= c1) return;                        // wave-uniform exit

    const int  m     = lane & 15;                // A: row M / B: col N
    const int  m8    = m & 7;                    // safe e column for all lanes
    const int  khalf = lane >> 4;                // lanes 16-31 carry K=2,3
    const int  r0    = 2 * khalf;                // this lane's row offset in chunk
    const bool arow  = (m < PDIM);               // rows 8..15 are zero padding

    const int  len4  = (c1 - c0) >> 2;           // full K=4 chunks
    const int  pairs = len4 >> 1;

    // Lane-local streaming pointers (advance by 4 rows per chunk).
    const float* ep = e + (size_t)(c0 + r0) * PDIM + m8;
    const float* xp = x + (size_t)(c0 + r0) * COLS + col0 + m;

    v8f acc0 = {}, acc1 = {};

    // ---- main loop: 2 chunks (8 nodes) per iteration, 2 indep. WMMA chains
    for (int i = 0; i < pairs; ++i) {
        const float a0 = ep[0];
        const float a1 = ep[PDIM];
        const float b0 = xp[0];
        const float b1 = xp[COLS];
        const float a2 = ep[4 * PDIM];
        const float a3 = ep[5 * PDIM];
        const float b2 = xp[4 * COLS];
        const float b3 = xp[5 * COLS];
        __builtin_prefetch(xp + 16 * COLS, 0, 3);

        v2f A0, B0, A1, B1;
        A0.x = arow ? a0 : 0.0f;  A0.y = arow ? a1 : 0.0f;
        B0.x = b0;                B0.y = b1;
        A1.x = arow ? a2 : 0.0f;  A1.y = arow ? a3 : 0.0f;
        B1.x = b2;                B1.y = b3;

        acc0 = wmma_f32(A0, B0, acc0);
        acc1 = wmma_f32(A1, B1, acc1);

        ep += 8 * PDIM;
        xp += 8 * COLS;
    }

    // ---- odd full chunk
    if (len4 & 1) {
        const float a0 = ep[0];
        const float a1 = ep[PDIM];
        const float b0 = xp[0];
        const float b1 = xp[COLS];
        v2f A, B;
        A.x = arow ? a0 : 0.0f;  A.y = arow ? a1 : 0.0f;
        B.x = b0;                B.y = b1;
        acc0 = wmma_f32(A, B, acc0);
    }

    // ---- masked tail (1..3 nodes), clamped addresses + value masking
    const int nt  = c0 + (len4 << 2);
    if (nt < c1) {
        const int na  = nt + r0;
        const int nb  = na + 1;
        const int naC = min(na, c1 - 1);
        const int nbC = min(nb, c1 - 1);
        const float a0 = e[(size_t)naC * PDIM + m8];
        const float a1 = e[(size_t)nbC * PDIM + m8];
        const float b0 = x[(size_t)naC * COLS + col0 + m];
        const float b1 = x[(size_t)nbC * COLS + col0 + m];
        v2f A, B;
        A.x = (arow && na < c1) ? a0 : 0.0f;
        A.y = (arow && nb < c1) ? a1 : 0.0f;
        B.x = (na < c1) ? b0 : 0.0f;
        B.y = (nb < c1) ? b1 : 0.0f;
        acc0 = wmma_f32(A, B, acc0);
    }

    const v8f acc = acc0 + acc1;

    // C/D layout: VGPR r, lanes 0-15 -> M=r ; lanes 16-31 -> M=r+8 (padding).
    if (lane < 16) {
        #pragma unroll
        for (int r = 0; r < PDIM; ++r)
            atomicAdd(&accum[(b * PDIM + r) * COLS + col0 + m], acc[r]);
    }
}

// -------- scale by softmax denominator, permute [b,p,t,f] -> [b,t,p,f] ------
__global__ void finalize_kernel(const float* __restrict__ accum,
                                const float* __restrict__ S,
                                float* __restrict__ out) {
    const int idx = blockIdx.x * blockDim.x + threadIdx.x;
    if (idx >= NSEG * TDIM * PDIM * FDIM) return;
    const int f = idx & (FDIM - 1);
    const int p = (idx >> 5) & (PDIM - 1);
    const int t = (idx >> 8) & (TDIM - 1);
    const int b = idx >> 11;
    const float s = S[b * PDIM + p];
    out[idx] = accum[(b * PDIM + p) * COLS + t * FDIM + f] / (s + 1e-16f);
}

extern "C" void kernel_launch(void* const* d_in, const int* in_sizes, int n_in,
                              void* d_out, int out_size, void* d_ws, size_t ws_size,
                              hipStream_t stream) {
    const float* pos = (const float*)d_in[0];
    const float* x   = (const float*)d_in[1];
    const int*   seg = (const int*)  d_in[2];
    const float* W1  = (const float*)d_in[3];
    const float* b1  = (const float*)d_in[4];
    const float* W2  = (const float*)d_in[5];
    const float* b2  = (const float*)d_in[6];
    float* out = (float*)d_out;
    const int N = in_sizes[2];           // seg has one entry per node

    // workspace carve-out
    char* ws = (char*)d_ws;
    size_t off = 0;
    float* e = (float*)(ws + off);  off += (size_t)N * PDIM * sizeof(float);
    off = (off + 255) & ~(size_t)255;
    unsigned* Mkey = (unsigned*)(ws + off); off += NSEG * PDIM * sizeof(unsigned);
    float* S = (float*)(ws + off);          off += NSEG * PDIM * sizeof(float);
    int* starts = (int*)(ws + off);         off += 64 * sizeof(int);
    off = (off + 255) & ~(size_t)255;
    float* accum = (float*)(ws + off);      off += (size_t)NSEG * PDIM * COLS * sizeof(float);

    init_kernel<<<(NSEG * PDIM * COLS + 255) / 256, 256, 0, stream>>>(accum, Mkey, S);
    seg_starts_kernel<<<1, 64, 0, stream>>>(seg, N, starts);

    const int nb = (N + 255) / 256;
    mlp_max_kernel<<<nb, 256, 0, stream>>>(pos, seg, W1, b1, W2, b2, e, Mkey, N);
    exp_sum_kernel<<<nb, 256, 0, stream>>>(seg, Mkey, e, S, N);

    gemm_wmma_kernel<<<dim3(NSEG, 2, 8), 256, 0, stream>>>(x, e, starts, accum);

    finalize_kernel<<<(NSEG * TDIM * PDIM * FDIM + 255) / 256, 256, 0, stream>>>(accum, S, out);
}